// MultiBoxLoss_60189671686265
// MI455X (gfx1250) — compile-verified
//
#include <hip/hip_runtime.h>

#define BATCH  32
#define NPRIOR 24564
#define NCLS   81
#define NOBJ   24
#define TILE   128
#define TILES  ((NPRIOR + TILE - 1) / TILE)   // 192
#define NBLK   (BATCH * TILES)                // 6144

typedef __attribute__((ext_vector_type(2))) float v2f;
typedef __attribute__((ext_vector_type(8))) float v8f;
typedef __attribute__((ext_vector_type(4))) int   v4i;

#if __has_builtin(__builtin_amdgcn_global_load_async_to_lds_b128) && \
    __has_builtin(__builtin_amdgcn_s_wait_asynccnt)
#define HAVE_ASYNC_LDS 1
typedef __attribute__((address_space(1))) v4i* gbl_v4i_ptr;
typedef __attribute__((address_space(3))) v4i* lds_v4i_ptr;
#endif

__device__ __forceinline__ float sl1(float x) {
    float a = fabsf(x);
    return a < 1.0f ? 0.5f * x * x : a - 0.5f;
}

// ---------------- init: zero the per-truth best-prior packs ----------------
__global__ void passInit(unsigned long long* __restrict__ pack) {
    int i = blockIdx.x * blockDim.x + threadIdx.x;
    if (i < BATCH * NOBJ) pack[i] = 0ull;
}

// ---------------- pass A: matching (jaccard, argmaxes) ----------------
__global__ __launch_bounds__(256) void passA(
    const float* __restrict__ priors, const float* __restrict__ targets,
    unsigned long long* __restrict__ pack,
    int* __restrict__ bti, float* __restrict__ btov)
{
    __shared__ float sTgt[NOBJ * 5];
    int t = threadIdx.x;
    int b = blockIdx.y;
    int p = blockIdx.x * 256 + t;
    if (t < NOBJ * 5) sTgt[t] = targets[b * NOBJ * 5 + t];
    __syncthreads();

    bool valid = p < NPRIOR;
    float xl = 0.f, yl = 0.f, xr = 0.f, yr = 0.f;
    if (valid) {
        float px = priors[p * 4 + 0], py = priors[p * 4 + 1];
        float pw = priors[p * 4 + 2], ph = priors[p * 4 + 3];
        xl = px - pw * 0.5f; yl = py - ph * 0.5f;
        xr = px + pw * 0.5f; yr = py + ph * 0.5f;
    }
    float ab = (xr - xl) * (yr - yl);

    float bestOv = -1.0f; int bestI = 0;
    int lane = t & 31;
    for (int o = 0; o < NOBJ; ++o) {
        float x1 = sTgt[o * 5 + 0], y1 = sTgt[o * 5 + 1];
        float x2 = sTgt[o * 5 + 2], y2 = sTgt[o * 5 + 3];
        float iw = fmaxf(fminf(xr, x2) - fmaxf(xl, x1), 0.f);
        float ih = fmaxf(fminf(yr, y2) - fmaxf(yl, y1), 0.f);
        float inter = iw * ih;
        float aa = (x2 - x1) * (y2 - y1);
        float iou = inter / (aa + ab - inter);
        if (valid && iou > bestOv) { bestOv = iou; bestI = o; }
        // packed key: high 32 = iou bits (iou >= 0 so int order == float order),
        // low 32 = ~p so ties pick the smallest prior index (first-max).
        unsigned long long key = valid
            ? (((unsigned long long)__float_as_uint(iou) << 32) |
               (unsigned long long)(0xFFFFFFFFu - (unsigned)p))
            : 0ull;
        // wave32 max-reduce (exact, order independent)
        for (int off = 16; off > 0; off >>= 1) {
            unsigned hiw = (unsigned)(key >> 32), low = (unsigned)key;
            unsigned hio = __shfl_xor(hiw, off);
            unsigned loo = __shfl_xor(low, off);
            unsigned long long other = ((unsigned long long)hio << 32) | loo;
            key = other > key ? other : key;
        }
        if (lane == 0) atomicMax(&pack[b * NOBJ + o], key);
    }
    if (valid) {
        size_t gp = (size_t)b * NPRIOR + p;
        btov[gp] = bestOv;
        bti[gp]  = bestI;
    }
}

// ---------------- pass B: per-truth override (sequential = JAX scatter) ----
__global__ void passB(const unsigned long long* __restrict__ pack,
                      int* __restrict__ bti, float* __restrict__ btov)
{
    int b = threadIdx.x;
    if (b >= BATCH) return;
    for (int o = 0; o < NOBJ; ++o) {
        unsigned long long key = pack[b * NOBJ + o];
        unsigned p = 0xFFFFFFFFu - (unsigned)(key & 0xFFFFFFFFull);
        size_t gp = (size_t)b * NPRIOR + p;
        btov[gp] = 2.0f;
        bti[gp]  = o;
    }
}

// ---------------- pass C: logsumexp (WMMA sum), conf/loc losses ----------
__global__ __launch_bounds__(256) void passC(
    const float* __restrict__ loc, const float* __restrict__ conf,
    const float* __restrict__ priors, const float* __restrict__ targets,
    const int* __restrict__ bti, const float* __restrict__ btov,
    float* __restrict__ mined,
    float* __restrict__ pl, float* __restrict__ pc, int* __restrict__ pnp)
{
    __shared__ float sConf[TILE * NCLS];   // 41472 B
    __shared__ float sTgt[NOBJ * 5];
    __shared__ float sSums[8 * 16];
    __shared__ float sRedF[256];
    __shared__ int   sRedI[256];

    int t = threadIdx.x;
    int tile = blockIdx.x, b = blockIdx.y;
    int p0 = tile * TILE;
    int nvalid = min(TILE, NPRIOR - p0);
    const float* crow = conf + ((size_t)b * NPRIOR + p0) * NCLS;

    if (t < NOBJ * 5) sTgt[t] = targets[b * NOBJ * 5 + t];
    int total = nvalid * NCLS;             // always a multiple of 4
#ifdef HAVE_ASYNC_LDS
    {
        // tile base is 16B aligned for every tile (41472B stride, 7958736B batch stride)
        int total4 = total >> 2;
        for (int i = t; i < (TILE * NCLS) / 4; i += 256) {
            if (i < total4) {
                __builtin_amdgcn_global_load_async_to_lds_b128(
                    (gbl_v4i_ptr)(crow + i * 4),
                    (lds_v4i_ptr)&sConf[i * 4],
                    0, 0);
            } else {
                sConf[i * 4 + 0] = 0.f; sConf[i * 4 + 1] = 0.f;
                sConf[i * 4 + 2] = 0.f; sConf[i * 4 + 3] = 0.f;
            }
        }
        __builtin_amdgcn_s_wait_asynccnt(0);   // per-wave: drain own async ops
    }
#else
    for (int i = t; i < TILE * NCLS; i += 256)
        sConf[i] = (i < total) ? crow[i] : 0.0f;
#endif
    __syncthreads();

    int w = t >> 5, lane = t & 31;
    int r = w * 16 + (lane & 15);     // prior row in tile (16 per wave)
    int p = p0 + r;
    bool half = lane >= 16;
    const float* rowp = &sConf[r * NCLS];

    // split-half max over classes, then combine with XOR-16 partner
    float mx = -3.0e38f;
    {
        int c0 = half ? 41 : 0, c1 = half ? NCLS : 41;
        for (int c = c0; c < c1; ++c) mx = fmaxf(mx, rowp[c]);
    }
    mx = fmaxf(mx, __shfl_xor(mx, 16));

    int koff = half ? 2 : 0;          // A 16x4 f32 layout: lanes>=16 hold K=2,3
    v8f accv = {0.f, 0.f, 0.f, 0.f, 0.f, 0.f, 0.f, 0.f};
#if __has_builtin(__builtin_amdgcn_wmma_f32_16x16x4_f32)
    v2f onesv = {1.0f, 1.0f};
    for (int kc = 0; kc < 21; ++kc) {
        int c = kc * 4 + koff;
        float e0 = (c     < NCLS) ? __expf(rowp[c]     - mx) : 0.0f;
        float e1 = (c + 1 < NCLS) ? __expf(rowp[c + 1] - mx) : 0.0f;
        v2f a = {e0, e1};
        // D[m,n] = sum_k A[m,k]*1 + C  -> row sums of exp, accumulated over k-chunks
        accv = __builtin_amdgcn_wmma_f32_16x16x4_f32(
            false, a, false, onesv, (short)0, accv, false, false);
    }
    if (lane == 0 || lane == 16) {
        int base = w * 16 + (half ? 8 : 0);
        #pragma unroll
        for (int j = 0; j < 8; ++j) sSums[base + j] = accv[j];
    }
#else
    {
        float ps = 0.f;
        for (int kc = 0; kc < 21; ++kc) {
            int c = kc * 4 + koff;
            if (c     < NCLS) ps += __expf(rowp[c]     - mx);
            if (c + 1 < NCLS) ps += __expf(rowp[c + 1] - mx);
        }
        ps += __shfl_xor(ps, 16);
        if (lane < 16) sSums[w * 16 + lane] = ps;
    }
#endif
    __syncthreads();

    float myL = 0.f, myC = 0.f; int myN = 0;
    if (lane < 16 && p < NPRIOR) {
        float sumexp = sSums[w * 16 + lane];
        float lse = mx + __logf(sumexp);
        size_t gp = (size_t)b * NPRIOR + p;
        int ti = bti[gp];
        float ov = btov[gp];
        int ct = 0;
        if (ov >= 0.5f) ct = (int)sTgt[ti * 5 + 4] + 1;
        float gathered = sConf[r * NCLS + ct];
        float lca = lse - gathered;
        bool pos = ct > 0;
        mined[gp] = pos ? 0.0f : lca;
        if (pos) {
            myC = lca; myN = 1;
            float px = priors[p * 4 + 0], py = priors[p * 4 + 1];
            float pw = priors[p * 4 + 2], ph = priors[p * 4 + 3];
            float x1 = sTgt[ti * 5 + 0], y1 = sTgt[ti * 5 + 1];
            float x2 = sTgt[ti * 5 + 2], y2 = sTgt[ti * 5 + 3];
            float gcx = ((x1 + x2) * 0.5f - px) / (0.1f * pw);
            float gcy = ((y1 + y2) * 0.5f - py) / (0.1f * ph);
            float gw  = __logf((x2 - x1) / pw) / 0.2f;
            float gh  = __logf((y2 - y1) / ph) / 0.2f;
            const float* ld = &loc[gp * 4];
            myL = sl1(ld[0] - gcx) + sl1(ld[1] - gcy)
                + sl1(ld[2] - gw ) + sl1(ld[3] - gh );
        }
    }

    // deterministic block tree reductions
    sRedF[t] = myL; __syncthreads();
    for (int s = 128; s > 0; s >>= 1) { if (t < s) sRedF[t] += sRedF[t + s]; __syncthreads(); }
    if (t == 0) pl[b * TILES + tile] = sRedF[0];
    __syncthreads();
    sRedF[t] = myC; __syncthreads();
    for (int s = 128; s > 0; s >>= 1) { if (t < s) sRedF[t] += sRedF[t + s]; __syncthreads(); }
    if (t == 0) pc[b * TILES + tile] = sRedF[0];
    __syncthreads();
    sRedI[t] = myN; __syncthreads();
    for (int s = 128; s > 0; s >>= 1) { if (t < s) sRedI[t] += sRedI[t + s]; __syncthreads(); }
    if (t == 0) pnp[b * TILES + tile] = sRedI[0];
}

// ---------------- pass D: exact top-k sum via bit-pattern binary search ----
// Stage the whole batch row (98KB) in LDS once; all 31 count iterations and
// the final sum read LDS instead of re-reading ~3MB/batch from L2.
__global__ __launch_bounds__(256) void passD(
    const float* __restrict__ mined, const int* __restrict__ pnp,
    float* __restrict__ topk)
{
    __shared__ float sMined[NPRIOR];      // 98256 B (<= 320KB/WGP on CDNA5)
    __shared__ int   sI[256];
    __shared__ float sF[256];

    int b = blockIdx.x, t = threadIdx.x;
    const float* row = mined + (size_t)b * NPRIOR;
    for (int i = t; i < NPRIOR; i += 256) sMined[i] = row[i];

    int npl = 0;
    for (int i = t; i < TILES; i += 256) npl += pnp[b * TILES + i];
    sI[t] = npl; __syncthreads();
    for (int s = 128; s > 0; s >>= 1) { if (t < s) sI[t] += sI[t + s]; __syncthreads(); }
    int np = sI[0]; __syncthreads();

    int k = min(3 * np, NPRIOR - 1);
    if (k <= 0) { if (t == 0) topk[b] = 0.0f; return; }

    // mined >= 0 -> uint bit order == float order. Find k-th largest value.
    unsigned lo = 0u, hi = 0x7F800000u;   // invariant: count(> hi) < k
    for (int it = 0; it < 31; ++it) {
        unsigned mid = (lo + hi) >> 1;
        int cnt = 0;
        for (int i = t; i < NPRIOR; i += 256)
            cnt += (__float_as_uint(sMined[i]) > mid) ? 1 : 0;
        sI[t] = cnt; __syncthreads();
        for (int s = 128; s > 0; s >>= 1) { if (t < s) sI[t] += sI[t + s]; __syncthreads(); }
        int c = sI[0]; __syncthreads();
        if (c >= k) lo = mid + 1; else hi = mid;
    }
    float tval = __uint_as_float(lo);     // k-th largest value
    int cgt = 0; float sgt = 0.f;
    for (int i = t; i < NPRIOR; i += 256) {
        float v = sMined[i];
        if (__float_as_uint(v) > lo) { cgt++; sgt += v; }
    }
    sI[t] = cgt; sF[t] = sgt; __syncthreads();
    for (int s = 128; s > 0; s >>= 1) {
        if (t < s) { sI[t] += sI[t + s]; sF[t] += sF[t + s]; }
        __syncthreads();
    }
    if (t == 0) topk[b] = sF[0] + (float)(k - sI[0]) * tval;
}

// ---------------- pass E: deterministic serial finalize ----------------
__global__ void passE(const float* __restrict__ pl, const float* __restrict__ pc,
                      const int* __restrict__ pnp, const float* __restrict__ topk,
                      float* __restrict__ out)
{
    if (threadIdx.x != 0 || blockIdx.x != 0) return;
    double L = 0.0, C = 0.0; long long N = 0;
    for (int i = 0; i < NBLK; ++i) { L += (double)pl[i]; C += (double)pc[i]; N += pnp[i]; }
    for (int b = 0; b < BATCH; ++b) C += (double)topk[b];
    double Nd = (double)N;
    out[0] = (float)(L / Nd);
    out[1] = (float)(C / Nd);
}

// ---------------- host launcher ----------------
extern "C" void kernel_launch(void* const* d_in, const int* in_sizes, int n_in,
                              void* d_out, int out_size, void* d_ws, size_t ws_size,
                              hipStream_t stream) {
    (void)in_sizes; (void)n_in; (void)out_size; (void)ws_size;
    const float* loc     = (const float*)d_in[0];
    const float* conf    = (const float*)d_in[1];
    const float* priors  = (const float*)d_in[2];
    const float* targets = (const float*)d_in[3];
    float* out = (float*)d_out;

    char* ws = (char*)d_ws;
    size_t off = 0;
    unsigned long long* pack = (unsigned long long*)(ws + off); off += (size_t)BATCH * NOBJ * 8;
    int*   bti  = (int*)  (ws + off); off += (size_t)BATCH * NPRIOR * 4;
    float* btov = (float*)(ws + off); off += (size_t)BATCH * NPRIOR * 4;
    float* mined= (float*)(ws + off); off += (size_t)BATCH * NPRIOR * 4;
    float* pl   = (float*)(ws + off); off += (size_t)NBLK * 4;
    float* pc   = (float*)(ws + off); off += (size_t)NBLK * 4;
    int*   pnp  = (int*)  (ws + off); off += (size_t)NBLK * 4;
    float* topk = (float*)(ws + off); off += (size_t)BATCH * 4;

    passInit<<<(BATCH * NOBJ + 255) / 256, 256, 0, stream>>>(pack);
    dim3 gA((NPRIOR + 255) / 256, BATCH);
    passA<<<gA, 256, 0, stream>>>(priors, targets, pack, bti, btov);
    passB<<<1, 32, 0, stream>>>(pack, bti, btov);
    dim3 gC(TILES, BATCH);
    passC<<<gC, 256, 0, stream>>>(loc, conf, priors, targets, bti, btov,
                                  mined, pl, pc, pnp);
    passD<<<BATCH, 256, 0, stream>>>(mined, pnp, topk);
    passE<<<1, 1, 0, stream>>>(pl, pc, pnp, topk, out);
}